// AudioMixer_5600637354315
// MI455X (gfx1250) — compile-verified
//
#include <hip/hip_runtime.h>
#include <math.h>

#define SRATE   48000
#define NSMP    1440000
#define NTRK    4
#define NCH     2

typedef __attribute__((ext_vector_type(2))) float v2f;
typedef __attribute__((ext_vector_type(8))) float v8f;

// ---- workspace layout (in floats) ----
#define OFF_L     0                       // 4 tracks * 16*16  (impulse Toeplitz, row-major)
#define OFF_G     1024                    // 4 * 16*8          (state->output, cols 6,7 unused)
#define OFF_P     1536                    // 4 * 8*16          (input->state over block)
#define OFF_A16   2048                    // 4 * 8*8           (A^16 state transition)
#define OFF_GAIN  2304                    // gL[4], gR[4]
#define OFF_BUFA  4096                    // post-EQ: 4*2*NSMP
#define OFF_BUFB  (4096 + NTRK*NCH*NSMP)  // post-compressor
#define WET_OFF   OFF_BUFA                // reuse bufA after compressor: wet 2*2*NSMP
#define AP1_OFF   (OFF_BUFA + 2*NCH*NSMP) // allpass1 out

// ---- EQ segmenting ----
#define SEG   2048
#define WARM  2048
#define NSEG  704                          // ceil(NSMP/SEG)
#define NGRP  44                           // NSEG/16 chains-per-wave groups
// ---- compressor segmenting ----
#define CSEG  1024
#define CWARM 32768
#define CNSEG 1407                         // ceil(NSMP/CSEG)

#define ATKC 0.99791889f
#define RELC 0.99979169f
#define THRC 0.12589254f
#define FBC  0.6f
#define CEILC 0.89125094f

// =====================================================================
// Setup: fold 3 biquads -> 6th-order IIR, build blocked state-space
// matrices (double precision), plus pan/volume gains.
// =====================================================================
__global__ void setup_kernel(const float* __restrict__ volumes,
                             const float* __restrict__ pans,
                             float* __restrict__ ws) {
  const int t = threadIdx.x;
  if (t < NTRK) {
    const double PRE[NTRK][3][3] = {
      {{300.0,-3.0,0.7},{3000.0,3.0,1.0},{8000.0,2.0,0.7}},   // voice
      {{80.0,  2.0,0.7},{5000.0,1.0,1.0},{0,0,0}},            // music
      {{200.0,-2.0,0.7},{6000.0,-1.0,0.7},{0,0,0}},           // ambient
      {{1000.0,2.0,1.0},{0,0,0},{0,0,0}},                     // fx
    };
    const int NB[NTRK] = {3,2,2,1};
    double b[7] = {1,0,0,0,0,0,0}, a[7] = {1,0,0,0,0,0,0};
    const double PI = 3.14159265358979323846;
    for (int s = 0; s < 3; s++) {
      double sb[3], sa[3];
      if (s < NB[t]) {
        double f = PRE[t][s][0], g = PRE[t][s][1], q = PRE[t][s][2];
        double A  = pow(10.0, g/40.0);
        double w0 = 2.0*PI*f/(double)SRATE;
        double al = sin(w0)/(2.0*q);
        double a0 = 1.0 + al/A;
        sb[0] = (1.0+al*A)/a0; sb[1] = -2.0*cos(w0)/a0; sb[2] = (1.0-al*A)/a0;
        sa[0] = 1.0;           sa[1] = -2.0*cos(w0)/a0; sa[2] = (1.0-al/A)/a0;
      } else { sb[0]=1; sb[1]=0; sb[2]=0; sa[0]=1; sa[1]=0; sa[2]=0; }
      double nb[7] = {0,0,0,0,0,0,0}, na[7] = {0,0,0,0,0,0,0};
      for (int i = 0; i < 7; i++)
        for (int j = 0; j < 3 && i + j < 7; j++) {
          nb[i+j] += b[i]*sb[j]; na[i+j] += a[i]*sa[j];
        }
      for (int i = 0; i < 7; i++) { b[i] = nb[i]; a[i] = na[i]; }
    }
    // TDF2: y = b0 x + s0 ; s_i' = s_{i+1} + b_{i+1} x - a_{i+1} y
    auto step = [&](double s[6], double x, double* yout) {
      double y = b[0]*x + s[0];
      double ns[6];
      for (int i = 0; i < 6; i++) {
        double znext = (i < 5) ? s[i+1] : 0.0;
        ns[i] = znext + b[i+1]*x - a[i+1]*y;
      }
      for (int i = 0; i < 6; i++) s[i] = ns[i];
      if (yout) *yout = y;
    };
    // impulse response -> L (lower-triangular Toeplitz)
    double s[6], h[16];
    for (int i = 0; i < 6; i++) s[i] = 0.0;
    for (int m = 0; m < 16; m++) { double y; step(s, (m==0)?1.0:0.0, &y); h[m] = y; }
    for (int m = 0; m < 16; m++)
      for (int k = 0; k < 16; k++)
        ws[OFF_L + t*256 + m*16 + k] = (m >= k) ? (float)h[m-k] : 0.0f;
    // G[m][j] = (A^m)[0][j]
    for (int j = 0; j < 6; j++) {
      for (int i = 0; i < 6; i++) s[i] = (i==j) ? 1.0 : 0.0;
      for (int m = 0; m < 16; m++) {
        ws[OFF_G + t*128 + m*8 + j] = (float)s[0];
        step(s, 0.0, nullptr);
      }
    }
    // P[:,k] = A^{15-k} B
    for (int k = 0; k < 16; k++) {
      for (int i = 0; i < 6; i++) s[i] = 0.0;
      for (int m = 0; m < 16; m++) step(s, (m==k)?1.0:0.0, nullptr);
      for (int i = 0; i < 6; i++) ws[OFF_P + t*128 + i*16 + k] = (float)s[i];
    }
    // A16
    for (int j = 0; j < 6; j++) {
      for (int i = 0; i < 6; i++) s[i] = (i==j) ? 1.0 : 0.0;
      for (int m = 0; m < 16; m++) step(s, 0.0, nullptr);
      for (int i = 0; i < 6; i++) ws[OFF_A16 + t*64 + i*8 + j] = (float)s[i];
    }
  }
  if (t == 0) {
    const double PI = 3.14159265358979323846;
    for (int k = 0; k < NTRK; k++) {
      double ang = ((double)pans[k] + 1.0) * 0.25 * PI;
      ws[OFF_GAIN + k]     = (float)((double)volumes[k] * cos(ang));
      ws[OFF_GAIN + 4 + k] = (float)((double)volumes[k] * sin(ang));
    }
  }
}

// =====================================================================
// EQ core: blocked 6th-order IIR via chained V_WMMA_F32_16X16X4_F32.
// One wave = 16 parallel segment-chains; per iteration one 16x16 output
// tile  Y = L*X + G*S  (6 WMMAs), scalar state update S' = A16*S + P*X.
// GUARD=false is the interior fast path: no bounds checks, b64 loads,
// two b128 stores, pointer-increment addressing.
// =====================================================================
template <bool GUARD>
__device__ __forceinline__ void eq_body(const float* __restrict__ x,
                                        float* __restrict__ y,
                                        long seg0, int h, int p,
                                        const v2f A[6],
                                        const float Pr[6][8],
                                        const float* __restrict__ Am) {
  const bool hi = (h == 1);
  float s0=0.f, s1=0.f, s2=0.f, s3=0.f, s4=0.f, s5=0.f;

  const float* px = x + (seg0 - WARM) + 2*h;  // row base for this lane
  float* py       = y + (seg0 - WARM) + 8*h;

  for (int blk = 0; blk < (SEG + WARM)/16; blk++) {
    const long base = seg0 - WARM + (long)blk*16;

    v2f B[6];
    if (GUARD) {
#pragma unroll
      for (int j = 0; j < 4; j++) {
        const long idx = base + 4*j + 2*h;
        v2f xv; xv.x = 0.0f; xv.y = 0.0f;
        if (idx >= 0 && idx + 1 < NSMP) xv = *(const v2f*)(x + idx);
        B[j] = xv;
      }
    } else {
      B[0] = *(const v2f*)(px + 0);
      B[1] = *(const v2f*)(px + 4);
      B[2] = *(const v2f*)(px + 8);
      B[3] = *(const v2f*)(px + 12);
      __builtin_prefetch((const void*)(px + 256), 0, 1);
    }
    B[4].x = hi ? s2 : s0;  B[4].y = hi ? s3 : s1;
    B[5].x = hi ? 0.f : s4; B[5].y = hi ? 0.f : s5;

    v8f c = {};
#pragma unroll
    for (int j = 0; j < 6; j++)
      c = __builtin_amdgcn_wmma_f32_16x16x4_f32(
            false, A[j], false, B[j], (short)0, c, false, false);

    if (blk >= WARM/16) {
      if (GUARD) {
        const long o = base + 8*h;
#pragma unroll
        for (int r = 0; r < 8; r++)
          if (o + r < NSMP) y[o + r] = c[r];
      } else {
        float4 lo4 = make_float4(c[0], c[1], c[2], c[3]);
        float4 hi4 = make_float4(c[4], c[5], c[6], c[7]);
        *(float4*)(py + 0) = lo4;
        *(float4*)(py + 4) = hi4;
      }
    }

    // S' = A16*S + P*X (per-chain, split across lane pair, combined via shfl)
    float fullp[6];
#pragma unroll
    for (int i = 0; i < 6; i++) {
      float pp = 0.0f;
#pragma unroll
      for (int j = 0; j < 4; j++)
        pp += Pr[i][2*j]*B[j].x + Pr[i][2*j+1]*B[j].y;
      pp += __shfl_xor(pp, 16, 32);
      fullp[i] = pp;
    }
    float ss[6] = {s0, s1, s2, s3, s4, s5};
    float ns[6];
#pragma unroll
    for (int i = 0; i < 6; i++) {
      float v = fullp[i];
#pragma unroll
      for (int j = 0; j < 6; j++) v += Am[i*8 + j]*ss[j];
      ns[i] = v;
    }
    s0 = ns[0]; s1 = ns[1]; s2 = ns[2]; s3 = ns[3]; s4 = ns[4]; s5 = ns[5];

    px += 16;
    py += 16;
  }
}

__global__ void __launch_bounds__(256) eq_kernel(const float* __restrict__ in,
                                                 float* __restrict__ ws) {
  const float* cst = ws;
  float* bufA = ws + OFF_BUFA;
  const int wid  = (blockIdx.x << 3) + (threadIdx.x >> 5);
  const int lane = threadIdx.x & 31;
  const int h = lane >> 4;          // half-wave (K/M split)
  const int p = lane & 15;          // A: output row M;  B/D: column (chain)
  const int t   = wid / (NCH*NGRP);
  const int rem = wid % (NCH*NGRP);
  const int ch  = rem / NGRP;
  const int grp = rem % NGRP;
  const long chain = (long)grp*16 + p;
  const long seg0  = chain * SEG;
  const long cbase = (long)(t*NCH + ch) * NSMP;
  const float* __restrict__ x = in + cbase;
  float* __restrict__ y = bufA + cbase;

  // A-operand: Aext = [ L(16x16) | G(16x6) | 0 ] as six 16x4 K-slices
  const float* Lm = cst + OFF_L + t*256;
  const float* Gm = cst + OFF_G + t*128;
  v2f A[6];
#pragma unroll
  for (int j = 0; j < 4; j++) {
    A[j].x = Lm[p*16 + 4*j + 2*h];
    A[j].y = Lm[p*16 + 4*j + 2*h + 1];
  }
  A[4].x = Gm[p*8 + 2*h]; A[4].y = Gm[p*8 + 2*h + 1];
  if (h == 0) { A[5].x = Gm[p*8 + 4]; A[5].y = Gm[p*8 + 5]; }
  else        { A[5].x = 0.0f;        A[5].y = 0.0f; }

  // P rows restricted to this lane's x-rows {4j+2h, 4j+2h+1}
  const float* Pm = cst + OFF_P + t*128;
  float Pr[6][8];
#pragma unroll
  for (int i = 0; i < 6; i++)
#pragma unroll
    for (int j = 0; j < 4; j++) {
      Pr[i][2*j]   = Pm[i*16 + 4*j + 2*h];
      Pr[i][2*j+1] = Pm[i*16 + 4*j + 2*h + 1];
    }
  const float* Am = cst + OFF_A16 + t*64;   // uniform -> scalar loads

  // Interior groups never touch n<0 (warm) nor n>=NSMP (tail): fast path.
  if (grp == 0 || grp == NGRP-1)
    eq_body<true >(x, y, seg0, h, p, A, Pr, Am);
  else
    eq_body<false>(x, y, seg0, h, p, A, Pr, Am);
}

// =====================================================================
// Compressor: segment-parallel envelope follower with warm-up.
// =====================================================================
__global__ void comp_kernel(float* __restrict__ ws) {
  const float* bufA = ws + OFF_BUFA;
  float* bufB = ws + OFF_BUFB;
  const int tid = blockIdx.x*256 + threadIdx.x;
  if (tid >= 8*CNSEG) return;
  const int seg = tid % CNSEG;
  const int c8  = tid / CNSEG;
  const long base = (long)c8 * NSMP;
  const int segstart = seg * CSEG;
  int start = segstart - CWARM; if (start < 0) start = 0;
  int end   = segstart + CSEG;  if (end > NSMP) end = NSMP;
  const float inv_thr = 1.0f / (THRC + 1e-8f);
  float env = 0.0f;
  for (int n = start; n < end; n++) {
    const float xv  = bufA[base + n];
    const float lvl = fabsf(xv);
    const float cc  = (lvl > env) ? ATKC : RELC;
    env = cc*env + (1.0f - cc)*lvl;
    if (n >= segstart) {
      const float gr = __powf(fmaxf(env * inv_thr, 1.0f), -0.75f);
      bufB[base + n] = xv * gr;
    }
  }
}

// =====================================================================
// Comb: y[n] = x[n] + fb*y[n-D]; D residue chains, 32-step warm-up.
// Accumulates wet += y/4; comb 0 initializes (launch order => determinism).
// =====================================================================
__global__ void comb_kernel(const float* __restrict__ bufB,
                            float* __restrict__ wet, int D, int first) {
  const int M = (NSMP + D - 1) / D;
  const int nsub = (M + 127) / 128;
  const int tid = blockIdx.x*256 + threadIdx.x;
  const long total = 4L * D * nsub;
  if (tid >= total) return;
  const int sub = tid % nsub;
  const int r   = (tid / nsub) % D;
  const int cc  = tid / (nsub * D);            // (tr2, ch) in 0..3
  const float* __restrict__ xin = bufB + (long)((2 + (cc >> 1))*2 + (cc & 1)) * NSMP;
  float* __restrict__ w = wet + (long)cc * NSMP;
  const int m0 = sub * 128;
  int ms = m0 - 32; if (ms < 0) ms = 0;
  int me = m0 + 128; if (me > M) me = M;
  float yv = 0.0f;
  for (int m = ms; m < me; m++) {
    const long idx = (long)m * D + r;
    if (idx >= NSMP) break;
    yv = xin[idx] + FBC * yv;
    if (m >= m0) {
      if (first) w[idx] = 0.25f * yv;
      else       w[idx] += 0.25f * yv;
    }
  }
}

// =====================================================================
// Allpass: y[n] = -fb x[n] + x[n-D] + fb y[n-D], y[0:D) = 0.
// =====================================================================
__global__ void ap_kernel(const float* __restrict__ in,
                          float* __restrict__ out, int D) {
  const int M = (NSMP + D - 1) / D;
  const int nsub = (M + 223) / 224;
  const int tid = blockIdx.x*256 + threadIdx.x;
  const long total = 4L * D * nsub;
  if (tid >= total) return;
  const int sub = tid % nsub;
  const int r   = (tid / nsub) % D;
  const int cc  = tid / (nsub * D);
  const float* __restrict__ xin = in + (long)cc * NSMP;
  float* __restrict__ yo = out + (long)cc * NSMP;
  const int m0 = sub * 224;
  int me = m0 + 224; if (me > M) me = M;
  float yv = 0.0f, xprev;
  int ms;
  if (sub == 0) {
    yo[r] = 0.0f;            // m = 0
    ms = 1; xprev = xin[r];
  } else {
    ms = m0 - 32;
    const long ip = (long)(ms - 1) * D + r;
    xprev = (ip < NSMP) ? xin[ip] : 0.0f;
  }
  for (int m = ms; m < me; m++) {
    const long idx = (long)m * D + r;
    const float xv = (idx < NSMP) ? xin[idx] : 0.0f;
    yv = -FBC * xv + xprev + FBC * yv;
    xprev = xv;
    if (m >= m0 && idx < NSMP) yo[idx] = yv;
  }
}

// =====================================================================
// Mix: out = clip(G * [dry | 0.7 dry + 0.3 wet]).
// =====================================================================
__global__ void mix_kernel(const float* __restrict__ ws, float* __restrict__ out) {
  const float* bufB = ws + OFF_BUFB;
  const float* wet2 = ws + WET_OFF;     // allpass-2 output
  const float* gain = ws + OFF_GAIN;
  const int n = blockIdx.x*256 + threadIdx.x;
  if (n >= NSMP) return;
  float vL = 0.0f, vR = 0.0f;
#pragma unroll
  for (int t = 0; t < 4; t++) {
    float xl = bufB[(long)(t*2 + 0)*NSMP + n];
    float xr = bufB[(long)(t*2 + 1)*NSMP + n];
    if (t >= 2) {
      const float wl = wet2[(long)((t-2)*2 + 0)*NSMP + n];
      const float wr = wet2[(long)((t-2)*2 + 1)*NSMP + n];
      xl = 0.7f*xl + 0.3f*wl;
      xr = 0.7f*xr + 0.3f*wr;
    }
    vL += gain[t]     * xl;
    vR += gain[4 + t] * xr;
  }
  out[n]        = fminf(fmaxf(vL, -CEILC), CEILC);
  out[NSMP + n] = fminf(fmaxf(vR, -CEILC), CEILC);
}

// =====================================================================
extern "C" void kernel_launch(void* const* d_in, const int* in_sizes, int n_in,
                              void* d_out, int out_size, void* d_ws, size_t ws_size,
                              hipStream_t stream) {
  const float* tracks  = (const float*)d_in[0];
  const float* volumes = (const float*)d_in[1];
  const float* pans    = (const float*)d_in[2];
  float* out = (float*)d_out;
  float* wsf = (float*)d_ws;
  float* bufB = wsf + OFF_BUFB;
  float* wet  = wsf + WET_OFF;
  float* ap1  = wsf + AP1_OFF;

  auto cdiv = [](long a, long b) { return (a + b - 1) / b; };

  setup_kernel<<<1, 64, 0, stream>>>(volumes, pans, wsf);

  // EQ: 352 waves = 44 blocks * 8 waves
  eq_kernel<<<NGRP, 256, 0, stream>>>(tracks, wsf);

  comp_kernel<<<(int)cdiv(8L*CNSEG, 256), 256, 0, stream>>>(wsf);

  const int combD[4] = {1440, 1627, 1828, 2030};
  for (int k = 0; k < 4; k++) {
    const int D = combD[k];
    const long M = cdiv(NSMP, D);
    const long nsub = cdiv(M, 128);
    const long total = 4L * D * nsub;
    comb_kernel<<<(int)cdiv(total, 256), 256, 0, stream>>>(bufB, wet, D, k == 0 ? 1 : 0);
  }
  {
    const int D = 240;
    const long M = cdiv(NSMP, D), nsub = cdiv(M, 224);
    ap_kernel<<<(int)cdiv(4L*D*nsub, 256), 256, 0, stream>>>(wet, ap1, D);
  }
  {
    const int D = 81;
    const long M = cdiv(NSMP, D), nsub = cdiv(M, 224);
    ap_kernel<<<(int)cdiv(4L*D*nsub, 256), 256, 0, stream>>>(ap1, wet, D);
  }

  mix_kernel<<<(int)cdiv(NSMP, 256), 256, 0, stream>>>(wsf, out);
}